// Ensemble_37125697306783
// MI455X (gfx1250) — compile-verified
//
#include <hip/hip_runtime.h>
#include <math.h>
#include <stdint.h>

#define NBOX 4096
#define NTHR 1024
#define MM 20      // max merged boxes
#define HH 16      // hidden dim

static constexpr float kCond = 0.2f;
static constexpr float kIou  = 0.3f;

typedef float v2f __attribute__((ext_vector_type(2)));
typedef float v8f __attribute__((ext_vector_type(8)));

__device__ __forceinline__ v8f wmma_f32_k4(v2f a, v2f b, v8f c) {
    // V_WMMA_F32_16X16X4_F32 : D = A(16x4) * B(4x16) + C, full fp32
    return __builtin_amdgcn_wmma_f32_16x16x4_f32(
        false, a, false, b, (short)0, c, false, false);
}

// LDS layout (104 KB total):
//   [0, 65536)        bev  float[4096*4]
//   [65536, 69632)    label u8[4096]
//   [69632, 73728)    supp  u8[4096]
//   [73728, 106496)   sort keys u64[4096]  (reused after sort as MLP scratch)
#define OFF_LAB   65536
#define OFF_SUP   69632
#define OFF_SCR   73728
#define SMEM_SZ   106496

__global__ __launch_bounds__(NTHR)
void nms_merge_kernel(const float* __restrict__ boxes,
                      const float* __restrict__ scores,
                      const int*   __restrict__ labels,
                      const float* __restrict__ w1, const float* __restrict__ b1,
                      const float* __restrict__ w2, const float* __restrict__ b2,
                      const float* __restrict__ w3, const float* __restrict__ b3,
                      float* __restrict__ out, float* __restrict__ ws) {
    __shared__ __attribute__((aligned(16))) unsigned char smem[SMEM_SZ];
    float*              bevL = (float*)smem;
    unsigned char*      labL = smem + OFF_LAB;
    unsigned char*      supL = smem + OFF_SUP;
    unsigned long long* keys = (unsigned long long*)(smem + OFF_SCR);
    // scratch aliases (valid only after sort+gather):
    float* xbuf   = (float*)(smem + OFF_SCR);               // [16][20] = 320 f
    float* hbuf   = (float*)(smem + OFF_SCR + 1280);        // [16][16] = 256 f
    float* nbox   = (float*)(smem + OFF_SCR + 2304);        // [8]
    int*   wsum   = (int*)  (smem + OFF_SCR + 2336);        // [32]
    int*   wexcl  = (int*)  (smem + OFF_SCR + 2464);        // [32]
    int*   cntTot = (int*)  (smem + OFF_SCR + 2592);        // [1]

    const int tid  = (int)threadIdx.x;
    const int lane = tid & 31;
    const int wid  = tid >> 5;

    // ---- Phase 0: build sort keys (descending score, stable by index) ----
    for (int r = tid; r < NBOX; r += NTHR) {
        float s = scores[r];
        float f = (s > kCond) ? s : -__builtin_inff();
        uint32_t u = __float_as_uint(f);
        u = (u & 0x80000000u) ? ~u : (u | 0x80000000u);  // monotone float->uint
        u = ~u;                                          // descending
        keys[r] = ((unsigned long long)u << 32) | (unsigned)r;
    }
    __syncthreads();

    // ---- Phase 1: bitonic sort (ascending composite key) ----
    for (unsigned k = 2; k <= NBOX; k <<= 1) {
        for (unsigned j = k >> 1; j > 0; j >>= 1) {
            for (unsigned t = (unsigned)tid; t < NBOX / 2; t += NTHR) {
                unsigned i1 = ((t & ~(j - 1)) << 1) | (t & (j - 1));
                unsigned i2 = i1 | j;
                unsigned long long a = keys[i1], b = keys[i2];
                bool up = ((i1 & k) == 0);
                if ((a > b) == up) { keys[i1] = b; keys[i2] = a; }
            }
            __syncthreads();
        }
    }

    // ---- Phase 2: gather sorted data; precompute BEV; emit scores/labels ----
    for (int r = tid; r < NBOX; r += NTHR) {
        unsigned src = (unsigned)(keys[r] & 0xffffffffu);
        float b9[9];
#pragma unroll
        for (int f = 0; f < 9; ++f) b9[f] = boxes[src * 9 + f];
        float sc = scores[src];
        int   lb = labels[src];
#pragma unroll
        for (int f = 0; f < 9; ++f) ws[r * 9 + f] = b9[f];
        const float PI = 3.14159265358979323846f;
        float ang = b9[6] - floorf(b9[6] / PI + 0.5f) * PI;
        bool  swp = fabsf(ang) >= (PI * 0.25f);
        float dx = swp ? b9[4] : b9[3];
        float dy = swp ? b9[3] : b9[4];
        bevL[r * 4 + 0] = b9[0] - dx * 0.5f;
        bevL[r * 4 + 1] = b9[1] - dy * 0.5f;
        bevL[r * 4 + 2] = b9[0] + dx * 0.5f;
        bevL[r * 4 + 3] = b9[1] + dy * 0.5f;
        labL[r] = (unsigned char)lb;
        supL[r] = (sc > kCond) ? 0 : 1;
        out[NBOX * 9 + r]        = sc;         // scores (sorted)
        out[NBOX * 9 + NBOX + r] = (float)lb;  // labels (sorted)
    }
    __syncthreads();

    // ---- Preload MLP weights into per-lane VGPRs (WMMA B-matrix layout) ----
    // B 4x16 fp32: vgpr r, lane l -> B[r + 2*(l>=16)][l&15]
    const int nn = lane & 15;
    const int kh = (lane >> 4) << 1;
    v2f rw1[5], rw2[4], rw3[4];
#pragma unroll
    for (int c = 0; c < 5; ++c) {
        int k0 = 4 * c + kh;
        v2f t; t.x = w1[k0 * HH + nn]; t.y = w1[(k0 + 1) * HH + nn];
        rw1[c] = t;
    }
#pragma unroll
    for (int c = 0; c < 4; ++c) {
        int k0 = 4 * c + kh;
        v2f t; t.x = w2[k0 * HH + nn]; t.y = w2[(k0 + 1) * HH + nn];
        rw2[c] = t;
        v2f t3; t3.x = 0.f; t3.y = 0.f;
        if (nn == 0) { t3.x = w3[k0]; t3.y = w3[k0 + 1]; }
        rw3[c] = t3;
    }
    const float b1n = b1[nn], b2n = b2[nn], b3v = b3[0];

    // ---- Phase 3: serial NMS-merge scan ----
    for (int i = 0; i < NBOX; ++i) {
        bool acti = (supL[i] == 0);          // uniform across block
        if (!acti) {                         // suppressed: passthrough row
            if (tid < 9) out[i * 9 + tid] = ws[i * 9 + tid];
            if (tid == 9) out[NBOX * 9 + 2 * NBOX + i] = 0.0f;  // keep=false
            continue;                        // uniform; no barriers skipped unevenly
        }
        if (tid < 320) xbuf[tid] = 0.0f;     // zero 16x20 A-matrix staging

        // --- IoU sweep: each thread owns 4 consecutive boxes ---
        int   li  = (int)labL[i];
        float ax0 = bevL[i * 4 + 0], ay0 = bevL[i * 4 + 1];
        float ax1 = bevL[i * 4 + 2], ay1 = bevL[i * 4 + 3];
        float areaA = (ax1 - ax0) * (ay1 - ay0);
        unsigned m4 = 0;
        int j0 = tid << 2;
#pragma unroll
        for (int c = 0; c < 4; ++c) {
            int j = j0 + c;
            if (supL[j] == 0 && (int)labL[j] == li) {
                float bx0 = bevL[j * 4 + 0], by0 = bevL[j * 4 + 1];
                float bx1 = bevL[j * 4 + 2], by1 = bevL[j * 4 + 3];
                float iw = fminf(ax1, bx1) - fmaxf(ax0, bx0);
                float ih = fminf(ay1, by1) - fmaxf(ay0, by0);
                float inter = fmaxf(iw, 0.f) * fmaxf(ih, 0.f);
                float areaB = (bx1 - bx0) * (by1 - by0);
                float iou = inter / fmaxf(areaA + areaB - inter, 1e-6f);
                if (iou > kIou) m4 |= (1u << c);
            }
        }
        // --- block-wide exclusive prefix sum of per-thread counts ---
        int cnt = __popc(m4);
        int inc = cnt;
#pragma unroll
        for (int d = 1; d < 32; d <<= 1) {
            int v = __shfl_up(inc, d, 32);
            if (lane >= d) inc += v;
        }
        if (lane == 31) wsum[wid] = inc;
        __syncthreads();
        if (wid == 0) {
            int w = wsum[lane];
            int winc = w;
#pragma unroll
            for (int d = 1; d < 32; d <<= 1) {
                int v = __shfl_up(winc, d, 32);
                if (lane >= d) winc += v;
            }
            wexcl[lane] = winc - w;
            if (lane == 31) *cntTot = winc;
        }
        __syncthreads();
        int base  = wexcl[wid] + (inc - cnt);
        int count = *cntTot;

        // --- scatter first 20 overlaps into x = ob^T (16x20) + suppress ---
#pragma unroll
        for (int c = 0; c < 4; ++c) {
            if ((m4 >> c) & 1u) {
                int j = j0 + c;
                if (base < MM) {
#pragma unroll
                    for (int f = 0; f < 7; ++f)
                        xbuf[f * MM + base] = ws[j * 9 + f];
                }
                base++;
                supL[j] = 1;                 // over & active
            }
        }
        __syncthreads();

        // --- mergenet via fp32 WMMA (wave 0, EXEC all-1s, uniform branch) ---
        if (count > 1 && wid == 0) {
            // layer 1: (16x20)@(20x16)+b1, relu ; 5 x WMMA K=4
            v8f acc = {0.f, 0.f, 0.f, 0.f, 0.f, 0.f, 0.f, 0.f};
#pragma unroll
            for (int c = 0; c < 5; ++c) {
                int k0 = 4 * c + kh;
                v2f a; a.x = xbuf[nn * MM + k0]; a.y = xbuf[nn * MM + k0 + 1];
                acc = wmma_f32_k4(a, rw1[c], acc);
            }
            const int hi = (lane >> 4) << 3;
#pragma unroll
            for (int r = 0; r < 8; ++r)
                hbuf[(r + hi) * HH + nn] = fmaxf(acc[r] + b1n, 0.f);

            // layer 2: (16x16)@(16x16)+b2, relu ; 4 x WMMA K=4
            v8f acc2 = {0.f, 0.f, 0.f, 0.f, 0.f, 0.f, 0.f, 0.f};
#pragma unroll
            for (int c = 0; c < 4; ++c) {
                int k0 = 4 * c + kh;
                v2f a; a.x = hbuf[nn * HH + k0]; a.y = hbuf[nn * HH + k0 + 1];
                acc2 = wmma_f32_k4(a, rw2[c], acc2);
            }
#pragma unroll
            for (int r = 0; r < 8; ++r)
                hbuf[(r + hi) * HH + nn] = fmaxf(acc2[r] + b2n, 0.f);

            // layer 3: (16x16)@(16x[1 pad 16]) ; 4 x WMMA K=4, col 0 is answer
            v8f acc3 = {0.f, 0.f, 0.f, 0.f, 0.f, 0.f, 0.f, 0.f};
#pragma unroll
            for (int c = 0; c < 4; ++c) {
                int k0 = 4 * c + kh;
                v2f a; a.x = hbuf[nn * HH + k0]; a.y = hbuf[nn * HH + k0 + 1];
                acc3 = wmma_f32_k4(a, rw3[c], acc3);
            }
            if (lane == 0) {
#pragma unroll
                for (int m = 0; m < 7; ++m) nbox[m] = acc3[m] + b3v;
            }
        }
        __syncthreads();

        // --- emit output row i ---
        if (tid < 7) {
            float v = (count > 1) ? nbox[tid] : ws[i * 9 + tid];
            if (tid >= 3 && tid <= 5) v = fmaxf(v, 1e-5f);   // dims clamp
            out[i * 9 + tid] = v;
        } else if (tid < 9) {
            out[i * 9 + tid] = ws[i * 9 + tid];               // extra feats
        }
        if (tid == 9) out[NBOX * 9 + 2 * NBOX + i] = 1.0f;    // keep=true
    }
}

extern "C" void kernel_launch(void* const* d_in, const int* in_sizes, int n_in,
                              void* d_out, int out_size, void* d_ws, size_t ws_size,
                              hipStream_t stream) {
    (void)in_sizes; (void)n_in; (void)out_size; (void)ws_size;
    const float* boxes  = (const float*)d_in[0];
    const float* scores = (const float*)d_in[1];
    const int*   labels = (const int*)d_in[2];
    const float* w1 = (const float*)d_in[3];
    const float* b1 = (const float*)d_in[4];
    const float* w2 = (const float*)d_in[5];
    const float* b2 = (const float*)d_in[6];
    const float* w3 = (const float*)d_in[7];
    const float* b3 = (const float*)d_in[8];
    float* out = (float*)d_out;
    float* ws  = (float*)d_ws;
    nms_merge_kernel<<<dim3(1), dim3(NTHR), 0, stream>>>(
        boxes, scores, labels, w1, b1, w2, b2, w3, b3, out, ws);
}